// ProteinGNNOutput_29326036697588
// MI455X (gfx1250) — compile-verified
//
#include <hip/hip_runtime.h>
#include <limits.h>

typedef float v2f __attribute__((ext_vector_type(2)));
typedef float v8f __attribute__((ext_vector_type(8)));

#define NB    16      // graphs (ptr has NB+1 entries)
#define IDMAX 4096    // shared id vocabulary
#define DD    256     // feature dim
#define KB    50      // K-splits for WMMA pooling (waves = 16*KB)

// ---------------- kernel 1: init first-occurrence table ----------------
__global__ void init_table_k(int* __restrict__ table, int n) {
    int i = blockIdx.x * blockDim.x + threadIdx.x;
    if (i < n) table[i] = INT_MAX;
}

// ---------------- kernel 2: build table: min local index per (graph,id) ----
__global__ void build_table_k(const int* __restrict__ node_index,
                              const int* __restrict__ ptr,
                              int* __restrict__ table, int n, int nb) {
    int i = blockIdx.x * blockDim.x + threadIdx.x;
    if (i >= n) return;
    int id = node_index[i];
    if (id < 0 || id >= IDMAX) return;
    int b = -1;
#pragma unroll 1
    for (int g = 0; g < nb; ++g) {
        if (ptr[g] <= i && i < ptr[g + 1]) { b = g; break; }
    }
    if (b < 0) return;                       // row outside [ptr[0], ptr[nb])
    int local = i - ptr[b];                  // first occurrence == min local pos
    atomicMin(&table[b * IDMAX + id], local);
}

// ---------------- kernel 3: sequence_outputs[b,l,:] = found ? x[local_pos,:] : 0
// 256 threads/block, 4 rows/block, one float4 per thread (D/4 == 64 chunks/row)
__global__ void seq_out_k(const int* __restrict__ input_ids,
                          const float4* __restrict__ x4,
                          const int* __restrict__ table,
                          float4* __restrict__ out4, int L) {
    int t   = threadIdx.x;
    int row = blockIdx.x * 4 + (t >> 6);     // global (b*L + l)
    int c   = t & 63;                        // float4 chunk in row
    int b   = row / L;
    int id  = input_ids[row];
    int pos = (id >= 0 && id < IDMAX) ? table[b * IDMAX + id] : INT_MAX;
    float4 v = make_float4(0.f, 0.f, 0.f, 0.f);
    if (pos != INT_MAX) v = x4[(size_t)pos * (DD / 4) + c];  // LOCAL index, faithful
    out4[(size_t)row * (DD / 4) + c] = v;
}

// ---------------- kernel 4: segment sum as WMMA f32 16x16x4 GEMM ------------
// D(16x16) += A(16x4: x^T feature-chunk) x B(4x16: graph one-hot indicator)
// M = feature offset within d-tile, N = graph (exactly 16), K = node rows.
// Unrolled x4 (K=16 per trip): 8 loads clause-batched ahead of 4 WMMAs.
__global__ void pool_wmma_k(const float* __restrict__ x,
                            const int* __restrict__ ptr,
                            float* __restrict__ part, int n, int chunk) {
    int dt   = blockIdx.x;       // d-tile 0..15
    int kb   = blockIdx.y;       // k-block 0..KB-1
    int lane = threadIdx.x;      // wave32
    int col  = lane & 15;        // M index (A) and N/graph index (B,D)
    int hi   = lane >> 4;        // lane-half -> K rows {0,1} vs {2,3}
    int d0   = dt * 16;

    int p_lo = ptr[col];         // this lane's graph bounds (for B indicator)
    int p_hi = ptr[col + 1];

    int k0 = kb * chunk;
    int k1 = k0 + chunk;
    if (k1 > n) k1 = n;

    v8f c = {0.f, 0.f, 0.f, 0.f, 0.f, 0.f, 0.f, 0.f};
    // chunk is a multiple of 16, so each trip's 16 rows never cross into the
    // next k-block; rows >= n get indicator 0 (ptr[NB] <= n) and clamped loads.
    for (int i0 = k0; i0 < k1; i0 += 16) {
        v2f a[4], bb[4];
#pragma unroll
        for (int s = 0; s < 4; ++s) {        // issue all 8 loads first
            int r0  = i0 + 4 * s + 2 * hi;
            int r1  = r0 + 1;
            int rr0 = r0 < n ? r0 : n - 1;
            int rr1 = r1 < n ? r1 : n - 1;
            a[s].x  = x[(size_t)rr0 * DD + d0 + col];
            a[s].y  = x[(size_t)rr1 * DD + d0 + col];
            bb[s].x = (r0 >= p_lo && r0 < p_hi) ? 1.0f : 0.0f;
            bb[s].y = (r1 >= p_lo && r1 < p_hi) ? 1.0f : 0.0f;
        }
#pragma unroll
        for (int s = 0; s < 4; ++s)
            c = __builtin_amdgcn_wmma_f32_16x16x4_f32(false, a[s], false, bb[s],
                                                      (short)0, c, false, false);
    }

    // D layout: VGPR j -> feature row (j + 8*hi), lane%16 -> graph
    float* dst = &part[(((size_t)dt * KB + kb) * 16 + col) * 16];
#pragma unroll
    for (int j = 0; j < 8; ++j) dst[j + 8 * hi] = c[j];
}

// ---------------- kernel 5: deterministic reduction of K-block partials -----
__global__ void reduce_pool_k(const float* __restrict__ part,
                              float* __restrict__ outg) {
    int idx = blockIdx.x * blockDim.x + threadIdx.x;   // 0..NB*DD-1
    int b  = idx / DD;
    int d  = idx % DD;
    int dt = d >> 4, dl = d & 15;
    float s = 0.f;
#pragma unroll 1
    for (int kb = 0; kb < KB; ++kb)
        s += part[(((size_t)dt * KB + kb) * 16 + b) * 16 + dl];
    outg[idx] = s;
}

extern "C" void kernel_launch(void* const* d_in, const int* in_sizes, int n_in,
                              void* d_out, int out_size, void* d_ws, size_t ws_size,
                              hipStream_t stream) {
    const int*   input_ids  = (const int*)d_in[0];   // [B, L] int32
    const int*   node_index = (const int*)d_in[1];   // [N]    int32
    const float* x          = (const float*)d_in[2]; // [N, D] f32
    const int*   ptr        = (const int*)d_in[3];   // [B+1]  int32

    const int rows = in_sizes[0];          // B*L
    const int n    = in_sizes[1];          // N
    const int nb   = in_sizes[3] - 1;      // B (== NB)
    const int L    = rows / nb;

    float* out      = (float*)d_out;
    float* out_seq  = out;                             // [B*L, D]
    float* out_grph = out + (size_t)rows * DD;         // [B, D]

    int*   table = (int*)d_ws;                                         // 256 KB
    float* part  = (float*)((char*)d_ws + (size_t)NB * IDMAX * sizeof(int)); // 800 KB

    // 1) table init
    {
        int cnt = NB * IDMAX;
        init_table_k<<<(cnt + 255) / 256, 256, 0, stream>>>(table, cnt);
    }
    // 2) table build (atomicMin -> order independent -> deterministic)
    build_table_k<<<(n + 255) / 256, 256, 0, stream>>>(node_index, ptr, table, n, nb);

    // 3) sequence outputs (bandwidth-dominant: ~33.5 MB write + L2-resident gathers)
    seq_out_k<<<rows / 4, 256, 0, stream>>>(input_ids, (const float4*)x, table,
                                            (float4*)out_seq, L);

    // 4) WMMA pooling partials (chunk forced to a multiple of 16 for the x4 unroll)
    int chunk = ((n + KB * 16 - 1) / (KB * 16)) * 16;
    pool_wmma_k<<<dim3(DD / 16, KB), 32, 0, stream>>>(x, ptr, part, n, chunk);

    // 5) deterministic reduce -> graph_outputs
    reduce_pool_k<<<(NB * DD) / 256, 256, 0, stream>>>(part, out_grph);
}